// TasteGNN_11759620456961
// MI455X (gfx1250) — compile-verified
//
#include <hip/hip_runtime.h>
#include <math.h>

// ---------------------------------------------------------------------------
// HAN-style single-relation GNN layer for MI455X (gfx1250, wave32, WMMA).
//   h_src = x_ing @ W_ing + b_ing            (WMMA f32 16x16x4 GEMM)
//   h_dst = x_taste @ W_taste + b_taste      (WMMA f32 16x16x4 GEMM)
//   alpha = leaky_relu(h_src@a_src [src] + h_dst@a_dst [dst])
//   scatter-softmax over dst, weighted scatter-add, ReLU, GELU(erf).
// Semantic attention over one relation is identically beta==1 -> skipped.
// ---------------------------------------------------------------------------

typedef __attribute__((ext_vector_type(2))) float v2f;
typedef __attribute__((ext_vector_type(8))) float v8f;

#define DIM  128
#define XPAD 132   // padded LDS row stride (floats): 16B-aligned rows, no bank conflicts

// --- WMMA GEMM: H[M x 128] = X[M x 128] @ W[128 x 128] + bias --------------
// Persistent blocks, grid-stride over 16-row tiles. Block = 256 thr = 8 waves;
// wave w computes the 16x16 tile at columns n0 = 16w. W (64 KB) + bias loaded
// to LDS once per block; each 16x128 X slab staged in LDS coalesced per tile.
__global__ void __launch_bounds__(256) proj_gemm(const float* __restrict__ X,
                                                 const float* __restrict__ W,
                                                 const float* __restrict__ bias,
                                                 float* __restrict__ H,
                                                 int ntiles) {
    __shared__ float Wl[DIM * DIM];     // 64 KB
    __shared__ float Xs[16 * XPAD];     // 8.25 KB, padded
    __shared__ float bl[DIM];

    const int tid = threadIdx.x;
    // Load W into LDS once: 4096 float4 / 256 threads, coalesced.
    const float4* Wv  = (const float4*)W;
    float4*       Wlv = (float4*)Wl;
#pragma unroll
    for (int i = 0; i < (DIM * DIM / 4) / 256; ++i)
        Wlv[tid + i * 256] = Wv[tid + i * 256];
    if (tid < DIM) bl[tid] = bias[tid];

    const int wave = tid >> 5;          // 0..7 -> column tile
    const int lane = tid & 31;
    const int half = lane >> 4;         // 0: K={0,1}, 1: K={2,3} per WMMA step
    const int r    = lane & 15;
    const int n0   = wave * 16;

    for (int t = blockIdx.x; t < ntiles; t += gridDim.x) {
        const int m0 = t * 16;
        __syncthreads();                // Xs consumers of previous tile done
        // Stage 16x128 X slab: 512 float4 / 256 threads, fully coalesced.
        const float4* Xv = (const float4*)(X + (size_t)m0 * DIM);
#pragma unroll
        for (int i = 0; i < 2; ++i) {
            const int idx = tid + i * 256;   // 0..511
            const int row = idx >> 5;        // 32 float4 per source row
            const int col = idx & 31;
            *(float4*)&Xs[row * XPAD + col * 4] = Xv[idx];
        }
        __syncthreads();

        // Prefetch next slab while this one computes (global_prefetch_b8).
        if (t + gridDim.x < ntiles) {
            const float* nx = X + (size_t)(t + gridDim.x) * 16 * DIM;
            __builtin_prefetch(nx + tid * 8, 0, 0);  // 256 thr x 32B = 8 KB
        }

        v8f c = {};
#pragma unroll 4
        for (int k = 0; k < DIM; k += 4) {
            const int kk = k + 2 * half;
            // A 16x4 tile from LDS: lane r holds X[m0+r][kk], X[m0+r][kk+1].
            v2f a;
            a.x = Xs[r * XPAD + kk];
            a.y = Xs[r * XPAD + kk + 1];
            // B 4x16 tile from LDS: lane r holds W[kk][n0+r], W[kk+1][n0+r].
            v2f b;
            b.x = Wl[kk * DIM + n0 + r];
            b.y = Wl[(kk + 1) * DIM + n0 + r];
            c = __builtin_amdgcn_wmma_f32_16x16x4_f32(
                    /*neg_a=*/false, a, /*neg_b=*/false, b,
                    /*c_mod=*/(short)0, c, /*reuse_a=*/false, /*reuse_b=*/false);
        }
        // C/D layout: v[j] -> row m0 + j + 8*half, col n0 + r. Coalesced.
#pragma unroll
        for (int j = 0; j < 8; ++j) {
            const int m = m0 + j + 8 * half;
            H[(size_t)m * DIM + n0 + r] = c[j] + bl[n0 + r];
        }
    }
}

// --- per-row dot: out[i] = H[i,:] . a  (one wave32 per row) ----------------
__global__ void __launch_bounds__(256) row_dot(const float* __restrict__ H,
                                               const float* __restrict__ a,
                                               float* __restrict__ out, int nrows) {
    const int gwave = (int)((blockIdx.x * blockDim.x + threadIdx.x) >> 5);
    const int lane  = threadIdx.x & 31;
    if (gwave >= nrows) return;
    const float4 v  = ((const float4*)(H + (size_t)gwave * DIM))[lane];
    const float4 av = ((const float4*)a)[lane];
    float p = v.x * av.x + v.y * av.y + v.z * av.z + v.w * av.w;
#pragma unroll
    for (int off = 16; off > 0; off >>= 1) p += __shfl_down(p, off, 32);
    if (lane == 0) out[gwave] = p;
}

// --- monotone uint key for exact float atomic max --------------------------
__device__ __forceinline__ unsigned fkey(float f) {
    unsigned u = __float_as_uint(f);
    return (u & 0x80000000u) ? ~u : (u | 0x80000000u);
}
__device__ __forceinline__ float fdec(unsigned k) {
    return (k & 0x80000000u) ? __uint_as_float(k & 0x7fffffffu)
                             : __uint_as_float(~k);
}

// --- edge logits + segment max --------------------------------------------
__global__ void __launch_bounds__(256) edge_logits(const int* __restrict__ src,
                                                   const int* __restrict__ dst,
                                                   const float* __restrict__ als,
                                                   const float* __restrict__ ald,
                                                   float* __restrict__ alpha,
                                                   unsigned* __restrict__ amax, int E) {
    const int e = blockIdx.x * blockDim.x + threadIdx.x;
    if (e >= E) return;
    const int d = dst[e];
    float a = als[src[e]] + ald[d];
    a = (a > 0.0f) ? a : 0.2f * a;              // leaky_relu(0.2)
    alpha[e] = a;
    atomicMax(&amax[d], fkey(a));               // keys init to 0 == smallest
}

// --- exp(alpha - max) + segment sum ---------------------------------------
__global__ void __launch_bounds__(256) edge_exp(const int* __restrict__ dst,
                                                float* __restrict__ alpha,
                                                const unsigned* __restrict__ amax,
                                                float* __restrict__ denom, int E) {
    const int e = blockIdx.x * blockDim.x + threadIdx.x;
    if (e >= E) return;
    const int d = dst[e];
    const float ev = expf(alpha[e] - fdec(amax[d]));
    alpha[e] = ev;                               // overwrite in place
    atomicAdd(&denom[d], ev);
}

// --- weighted scatter-add aggregation: one wave32 per edge -----------------
// Lane l handles dims [4l, 4l+4): float4 gather of h_src row (coalesced 512B),
// 4 f32 atomic adds into the L2-resident output accumulator.
__global__ void __launch_bounds__(256) edge_aggregate(const int* __restrict__ src,
                                                      const int* __restrict__ dst,
                                                      const float* __restrict__ ev,
                                                      const float* __restrict__ denom,
                                                      const float* __restrict__ H,
                                                      float* __restrict__ out, int E) {
    const long long gid = (long long)blockIdx.x * blockDim.x + threadIdx.x;
    const int e    = (int)(gid >> 5);
    const int lane = (int)(gid & 31);
    if (e >= E) return;
    const int s = src[e];
    const int d = dst[e];
    const float attn = ev[e] / fmaxf(denom[d], 1e-16f);
    const float4 h = ((const float4*)(H + (size_t)s * DIM))[lane];
    float* o = out + (size_t)d * DIM + lane * 4;
    atomicAdd(o + 0, h.x * attn);
    atomicAdd(o + 1, h.y * attn);
    atomicAdd(o + 2, h.z * attn);
    atomicAdd(o + 3, h.w * attn);
}

// --- ReLU then exact-erf GELU: gelu(relu(x)) == x>0 ? gelu(x) : 0 ----------
__global__ void __launch_bounds__(256) finalize(float* __restrict__ out, int n) {
    const int i = blockIdx.x * blockDim.x + threadIdx.x;
    if (i >= n) return;
    const float x = out[i];
    out[i] = (x > 0.0f) ? 0.5f * x * (1.0f + erff(x * 0.70710678118654752f)) : 0.0f;
}

extern "C" void kernel_launch(void* const* d_in, const int* in_sizes, int n_in,
                              void* d_out, int out_size, void* d_ws, size_t ws_size,
                              hipStream_t stream) {
    const float* x_ing   = (const float*)d_in[0];
    const float* x_taste = (const float*)d_in[1];
    const int*   ei      = (const int*)d_in[2];
    const float* W_ing   = (const float*)d_in[3];
    const float* b_ing   = (const float*)d_in[4];
    const float* W_taste = (const float*)d_in[5];
    const float* b_taste = (const float*)d_in[6];
    const float* a_src   = (const float*)d_in[7];
    const float* a_dst   = (const float*)d_in[8];
    // d_in[9..11] (Wk, bk, q): semantic attention over one relation -> beta==1,
    // mathematically a no-op; intentionally unused.
    float* out = (float*)d_out;

    const int Ni = in_sizes[0] / DIM;   // 50000
    const int Nt = in_sizes[1] / DIM;   // 20000
    const int E  = in_sizes[2] / 2;     // 600000
    const int* src = ei;
    const int* dst = ei + E;

    // Workspace carve-out (~38.6 MB total).
    float* ws     = (float*)d_ws;
    float* h_src  = ws;                 ws += (size_t)Ni * DIM;
    float* h_dst  = ws;                 ws += (size_t)Nt * DIM;
    float* als    = ws;                 ws += Ni;
    float* ald    = ws;                 ws += Nt;
    float* alpha  = ws;                 ws += E;
    unsigned* amax = (unsigned*)ws;     ws += Nt;
    float* denom  = ws;                 ws += Nt;

    // Graph-capturable zero-init (amax key 0 == minimum possible key).
    hipMemsetAsync(amax,  0, (size_t)Nt * sizeof(unsigned), stream);
    hipMemsetAsync(denom, 0, (size_t)Nt * sizeof(float),    stream);
    hipMemsetAsync(out,   0, (size_t)Nt * DIM * sizeof(float), stream);

    // ~5 row-tiles per persistent block: amortizes the 64 KB W LDS stage.
    const int tiles_src = Ni / 16;                  // 3125
    const int tiles_dst = Nt / 16;                  // 1250
    const int grid_src  = (tiles_src + 4) / 5;      // 625
    const int grid_dst  = (tiles_dst + 4) / 5;      // 250
    proj_gemm<<<grid_src, 256, 0, stream>>>(x_ing,   W_ing,   b_ing,   h_src, tiles_src);
    proj_gemm<<<grid_dst, 256, 0, stream>>>(x_taste, W_taste, b_taste, h_dst, tiles_dst);

    row_dot<<<(Ni + 7) / 8, 256, 0, stream>>>(h_src, a_src, als, Ni);
    row_dot<<<(Nt + 7) / 8, 256, 0, stream>>>(h_dst, a_dst, ald, Nt);

    edge_logits<<<(E + 255) / 256, 256, 0, stream>>>(src, dst, als, ald, alpha, amax, E);
    edge_exp   <<<(E + 255) / 256, 256, 0, stream>>>(dst, alpha, amax, denom, E);

    const long long aggThreads = (long long)E * 32;
    edge_aggregate<<<(int)((aggThreads + 255) / 256), 256, 0, stream>>>(
        src, dst, alpha, denom, h_src, out, E);

    finalize<<<(Nt * DIM + 255) / 256, 256, 0, stream>>>(out, Nt * DIM);
}